// v_PT_70291434766890
// MI455X (gfx1250) — compile-verified
//
#include <hip/hip_runtime.h>
#include <stdint.h>

// ---------------------------------------------------------------------------
// v_PT gauge-covariant path transport, fused single pass.
// Lattice (16,16,16,32), spin 4, color 3, 16 paths.
// out[p](x) = M_p(x) * v_p(x + shift_p), with M_p a product of <=2 link
// matrices (possibly transposed) read at path-dependent shifted sites.
// Memory-bound (~210 MB @ 23.3 TB/s ~ 9us); staged via TDM async tensor
// loads into LDS (all source regions are contiguous T-lines).
// ---------------------------------------------------------------------------

typedef unsigned int u32x4 __attribute__((ext_vector_type(4)));
typedef int          i32x8 __attribute__((ext_vector_type(8)));
typedef int          i32x4 __attribute__((ext_vector_type(4)));

#if __has_builtin(__builtin_amdgcn_tensor_load_to_lds) && \
    __has_builtin(__builtin_amdgcn_s_wait_tensorcnt)
#define VPT_USE_TDM 1
#else
#define VPT_USE_TDM 0
#endif

// ---- per-path constant tables ------------------------------------------------
// nm: number of accumulated link matrices (0,1,2)
__constant__ signed char c_nm[16] = {0,1,1,1,1,1,1,1,1,2,2,2,2,2,2,2};
// mu of mat0 (applied first, i.e. leftmost) and mat1
__constant__ signed char c_mu[16][2] = {
  {0,0},{0,0},{0,0},{1,0},{1,0},{2,0},{2,0},{3,0},{3,0},
  {1,0},{2,1},{3,2},{1,0},{2,1},{3,2},{0,0}};
// transpose flags
__constant__ signed char c_tr[16][2] = {
  {0,0},{0,0},{1,0},{0,0},{1,0},{0,0},{1,0},{0,0},{1,0},
  {0,0},{0,0},{0,0},{0,1},{0,1},{0,1},{0,0}};
// site shifts (x,y,z,t) where each matrix is read
__constant__ signed char c_ms[16][2][4] = {
  {{0,0,0,0},{0,0,0,0}},   // p0  []
  {{0,0,0,0},{0,0,0,0}},   // p1  (0,+)
  {{-1,0,0,0},{0,0,0,0}},  // p2  (0,-)
  {{0,0,0,0},{0,0,0,0}},   // p3  (1,+)
  {{0,-1,0,0},{0,0,0,0}},  // p4  (1,-)
  {{0,0,0,0},{0,0,0,0}},   // p5  (2,+)
  {{0,0,-1,0},{0,0,0,0}},  // p6  (2,-)
  {{0,0,0,0},{0,0,0,0}},   // p7  (3,+)
  {{0,0,0,-1},{0,0,0,0}},  // p8  (3,-)
  {{0,0,0,0},{0,1,0,0}},   // p9  (0,+),(1,+)
  {{0,0,0,0},{0,0,1,0}},   // p10 (1,+),(2,+)
  {{0,0,0,0},{0,0,0,1}},   // p11 (2,+),(3,+)
  {{0,0,0,0},{-1,1,0,0}},  // p12 (0,-),(1,+)
  {{0,0,0,0},{0,-1,1,0}},  // p13 (1,-),(2,+)
  {{0,0,0,0},{0,0,-1,1}},  // p14 (2,-),(3,+)
  {{0,0,0,0},{1,0,0,0}},   // p15 (0,+2)
};
// shift at which v is gathered
__constant__ signed char c_vs[16][4] = {
  {0,0,0,0},{1,0,0,0},{-1,0,0,0},{0,1,0,0},{0,-1,0,0},
  {0,0,1,0},{0,0,-1,0},{0,0,0,1},{0,0,0,-1},
  {1,1,0,0},{0,1,1,0},{0,0,1,1},{-1,1,0,0},{0,-1,1,0},{0,0,-1,1},{2,0,0,0}};

#if VPT_USE_TDM
// 1-D contiguous DMA: ndw dwords from global `src` to LDS byte address
// `lds_addr` (low 32 bits of a flat LDS pointer == LDS address per ISA 10.2).
// D# bitfields per CDNA5 ISA §8.3-8.6: count=1, type=2(image), data_size=4B,
// tile_dim0 = tensor_dim0 = ndw, all higher dims collapsed to 1.
// This toolchain's builtin takes 6 args: (g0 v4u, g1 v8i, g2 v4i, g3 v4i,
// extra v8i, cpol) -- the trailing v8i is passed zero-filled (the probe-
// verified call shape on the clang-23 lane).
__device__ __forceinline__ void tdm_line_to_lds(uint32_t lds_addr,
                                                const float* src,
                                                uint32_t ndw) {
  uint64_t ga = (uint64_t)(uintptr_t)src;
  u32x4 g0 = { 1u,                                   // count=1, user descriptor
               lds_addr,                             // LDS byte address
               (uint32_t)ga,                         // global_addr[31:0]
               ((uint32_t)(ga >> 32) & 0x01FFFFFFu) | (2u << 30) }; // [56:32]|type=2
  i32x8 g1 = { (int)(2u << 16),                      // wg_mask=0, data_size=4B
               (int)((ndw & 0xFFFFu) << 16),         // tensor_dim0[15:0]
               (int)((ndw >> 16) | (1u << 16)),      // tensor_dim0[31:16] | tensor_dim1=1
               (int)(ndw << 16),                     // tile_dim0 = ndw
               1,                                    // tile_dim1=1, tile_dim2=0
               (int)ndw,                             // tensor_dim0_stride[31:0]
               (int)((ndw & 0xFFFFu) << 16),         // stride0 hi=0 | stride1 lo
               0 };                                  // stride1 hi
  i32x4 g2 = { 1, 1, (int)ndw, 0 };                  // dim2=1, dim3=1, dim2_stride, tile_dim3=0
  i32x4 g3 = { (int)ndw, (int)(1u << 16), 0, 0 };    // dim3_stride, tensor_dim4=1, tile_dim4=0
  i32x8 gz = { 0, 0, 0, 0, 0, 0, 0, 0 };             // unused trailing group
  __builtin_amdgcn_tensor_load_to_lds(g0, g1, g2, g3, gz, 0);
}
#endif

// Block = 128 threads = 4 waves; each wave handles one (path, x, y, z) T-line.
// LDS slice per wave: v line 384 dw | U0 line 288 dw | U1 line 288 dw = 960 dw.
__global__ __launch_bounds__(128) void v_pt_transport_kernel(
    const float* __restrict__ feats,   // [16][16][16][16][32][4][3]
    const float* __restrict__ U,       // [4][16][16][16][32][3][3]
    float* __restrict__ out) {         // [16][16][16][16][32][4][3]
  __shared__ float lds[4 * 960];

  const int wave = __builtin_amdgcn_readfirstlane((int)(threadIdx.x >> 5));
  const int lane = (int)(threadIdx.x & 31);          // == t coordinate
  const int task = (int)blockIdx.x * 4 + wave;       // 16 paths * 4096 lines
  const int p    = task >> 12;
  const int line = task & 4095;
  const int x  = line >> 8;
  const int yy = (line >> 4) & 15;
  const int zz = line & 15;

  const int nm = c_nm[p];
  float* slice = &lds[wave * 960];

  // ---- wave-uniform source line pointers (wrapped coordinates) ----
  int sx = (x + c_vs[p][0]) & 15;
  int sy = (yy + c_vs[p][1]) & 15;
  int sz = (zz + c_vs[p][2]) & 15;
  const float* vsrc =
      feats + ((size_t)p * 131072u + (size_t)((sx * 16 + sy) * 16 + sz) * 32u) * 12u;

  const float* m0src = nullptr;
  const float* m1src = nullptr;
  if (nm >= 1) {
    int ax = (x + c_ms[p][0][0]) & 15;
    int ay = (yy + c_ms[p][0][1]) & 15;
    int az = (zz + c_ms[p][0][2]) & 15;
    m0src = U + ((size_t)c_mu[p][0] * 131072u +
                 (size_t)((ax * 16 + ay) * 16 + az) * 32u) * 9u;
  }
  if (nm == 2) {
    int bx = (x + c_ms[p][1][0]) & 15;
    int by = (yy + c_ms[p][1][1]) & 15;
    int bz = (zz + c_ms[p][1][2]) & 15;
    m1src = U + ((size_t)c_mu[p][1] * 131072u +
                 (size_t)((bx * 16 + by) * 16 + bz) * 32u) * 9u;
  }

  // ---- stage contiguous T-lines into LDS ----
#if VPT_USE_TDM
  {
    uint32_t lb = (uint32_t)(uintptr_t)slice;        // flat low 32 == LDS addr
    tdm_line_to_lds(lb, vsrc, 384u);                 // 32 sites * 12 floats
    if (nm >= 1) tdm_line_to_lds(lb + 1536u, m0src, 288u); // 32 * 9 floats
    if (nm == 2) tdm_line_to_lds(lb + 2688u, m1src, 288u);
  }
  __builtin_amdgcn_s_wait_tensorcnt(0);
#else
  for (int i = lane; i < 384; i += 32) slice[i] = vsrc[i];
  if (nm >= 1)
    for (int i = lane; i < 288; i += 32) slice[384 + i] = m0src[i];
  if (nm == 2)
    for (int i = lane; i < 288; i += 32) slice[672 + i] = m1src[i];
  __syncthreads();   // reached by all threads (branches are wave-uniform)
#endif

  // ---- gather v with T-rotation ----
  const int tv = (lane + c_vs[p][3]) & 31;
  const float4* v4 = (const float4*)(slice + tv * 12);
  float4 va = v4[0], vb = v4[1], vc = v4[2];
  float vv[12] = {va.x, va.y, va.z, va.w, vb.x, vb.y, vb.z, vb.w,
                  vc.x, vc.y, vc.z, vc.w};

  // ---- accumulated 3x3 color matrix ----
  float M[3][3];
  if (nm == 0) {
    M[0][0] = 1.f; M[0][1] = 0.f; M[0][2] = 0.f;
    M[1][0] = 0.f; M[1][1] = 1.f; M[1][2] = 0.f;
    M[2][0] = 0.f; M[2][1] = 0.f; M[2][2] = 1.f;
  } else {
    const int t0 = (lane + c_ms[p][0][3]) & 31;
    const float* m0 = slice + 384 + t0 * 9;
    const bool tr0 = (bool)c_tr[p][0];
    float A0[3][3];
#pragma unroll
    for (int i = 0; i < 3; ++i)
#pragma unroll
      for (int j = 0; j < 3; ++j)
        A0[i][j] = tr0 ? m0[j * 3 + i] : m0[i * 3 + j];
    if (nm == 1) {
#pragma unroll
      for (int i = 0; i < 3; ++i)
#pragma unroll
        for (int j = 0; j < 3; ++j) M[i][j] = A0[i][j];
    } else {
      const int t1 = (lane + c_ms[p][1][3]) & 31;
      const float* m1 = slice + 672 + t1 * 9;
      const bool tr1 = (bool)c_tr[p][1];
      float A1[3][3];
#pragma unroll
      for (int i = 0; i < 3; ++i)
#pragma unroll
        for (int j = 0; j < 3; ++j)
          A1[i][j] = tr1 ? m1[j * 3 + i] : m1[i * 3 + j];
#pragma unroll
      for (int i = 0; i < 3; ++i)
#pragma unroll
        for (int j = 0; j < 3; ++j)
          M[i][j] = fmaf(A0[i][0], A1[0][j],
                    fmaf(A0[i][1], A1[1][j], A0[i][2] * A1[2][j]));
    }
  }

  // ---- apply to 4 spins and store ----
  float o[12];
#pragma unroll
  for (int s = 0; s < 4; ++s)
#pragma unroll
    for (int i = 0; i < 3; ++i)
      o[s * 3 + i] = fmaf(M[i][0], vv[s * 3 + 0],
                     fmaf(M[i][1], vv[s * 3 + 1], M[i][2] * vv[s * 3 + 2]));

  float4* op = (float4*)(out + ((size_t)p * 131072u +
                                (size_t)line * 32u + (size_t)lane) * 12u);
  op[0] = make_float4(o[0], o[1], o[2], o[3]);
  op[1] = make_float4(o[4], o[5], o[6], o[7]);
  op[2] = make_float4(o[8], o[9], o[10], o[11]);
}

extern "C" void kernel_launch(void* const* d_in, const int* in_sizes, int n_in,
                              void* d_out, int out_size, void* d_ws,
                              size_t ws_size, hipStream_t stream) {
  (void)in_sizes; (void)n_in; (void)out_size; (void)d_ws; (void)ws_size;
  const float* feats = (const float*)d_in[0];
  const float* U     = (const float*)d_in[1];
  float* out         = (float*)d_out;
  // 16 paths * 4096 lines = 65536 wave-tasks; 4 waves per 128-thread block.
  v_pt_transport_kernel<<<dim3(16384), dim3(128), 0, stream>>>(feats, U, out);
}